// Embedding_5738076307686
// MI455X (gfx1250) — compile-verified
//
#include <hip/hip_runtime.h>

// Embedding gather: out[r, :] = embeddings[x[r], :]
//   x:          [4, 2048]  int32   (8192 rows)
//   embeddings: [32000, 1024] f32  (128 MB table)
//   out:        [8192, 1024] f32   (32 MB)
//
// Pure bandwidth problem (~64 MB of HBM traffic -> ~2.7 us at 23.3 TB/s).
// We use the CDNA5 async global<->LDS DMA path (ASYNCcnt) so the data never
// touches VGPRs, and non-temporal stores so the write-once output does not
// evict the L2-resident embedding table.

#define EMBED_DIM        1024
#define CHUNKS           8            // 1024 floats / (32 lanes * 4 floats)
#define WAVES_PER_BLOCK  8
#define THREADS          (WAVES_PER_BLOCK * 32)

__global__ __launch_bounds__(THREADS)
void embed_gather_async(const int* __restrict__ x,
                        const float* __restrict__ emb,
                        float* __restrict__ out,
                        int rows)
{
    // Wave-private staging buffers: 8 waves * 4 KB = 32 KB LDS per block.
    __shared__ float lds[WAVES_PER_BLOCK][EMBED_DIM];

    const int lane       = threadIdx.x & 31;
    const int wave       = threadIdx.x >> 5;
    const int waveGlobal = blockIdx.x * WAVES_PER_BLOCK + wave;
    const int waveStride = gridDim.x * WAVES_PER_BLOCK;

    // 32-bit LDS byte address of this lane's 16 B slot in this wave's buffer.
    // (Shared-aperture flat address keeps the LDS offset in the low 32 bits.)
    unsigned ldsLane =
        (unsigned)(unsigned long long)(&lds[wave][0]) + (unsigned)lane * 16u;

    for (int row = waveGlobal; row < rows; row += waveStride) {
        const int    idx = x[row];
        const float* src = emb + (size_t)idx * EMBED_DIM;
        float*       dst = out + (size_t)row * EMBED_DIM;

        // Stage the 4 KB row global -> LDS. Each instruction moves
        // 32 lanes x 16 B = 512 B; tracked on ASYNCcnt, no VGPR data path.
#pragma unroll
        for (int i = 0; i < CHUNKS; ++i) {
            unsigned long long g =
                (unsigned long long)(src + i * 128 + lane * 4);
            unsigned l = ldsLane + (unsigned)i * 512u;
            asm volatile("global_load_async_to_lds_b128 %0, %1, off"
                         :
                         : "v"(l), "v"(g)
                         : "memory");
        }
        asm volatile("s_wait_asynccnt 0" ::: "memory");

        // Stream LDS -> global with non-temporal stores (write-once output;
        // keep L2 capacity for the 128 MB embedding table).
#pragma unroll
        for (int i = 0; i < CHUNKS; ++i) {
            unsigned long long g =
                (unsigned long long)(dst + i * 128 + lane * 4);
            unsigned l = ldsLane + (unsigned)i * 512u;
            asm volatile("global_store_async_from_lds_b128 %0, %1, off th:TH_STORE_NT"
                         :
                         : "v"(g), "v"(l)
                         : "memory");
        }
        // Buffer is reused next iteration: wait until stores consumed it.
        asm volatile("s_wait_asynccnt 0" ::: "memory");
    }
}

extern "C" void kernel_launch(void* const* d_in, const int* in_sizes, int n_in,
                              void* d_out, int out_size, void* d_ws, size_t ws_size,
                              hipStream_t stream)
{
    const int*   x   = (const int*)d_in[0];    // [8192] int32 indices
    const float* emb = (const float*)d_in[1];  // [32000, 1024] f32
    float*       out = (float*)d_out;          // [8192, 1024] f32

    const int rows   = in_sizes[0];                                   // 8192
    const int blocks = (rows + WAVES_PER_BLOCK - 1) / WAVES_PER_BLOCK; // 1024

    embed_gather_async<<<blocks, THREADS, 0, stream>>>(x, emb, out, rows);
}